// LossFunc_16338055594676
// MI455X (gfx1250) — compile-verified
//
#include <hip/hip_runtime.h>
#include <math.h>

// ---------------------------------------------------------------------------
// Problem constants (from reference)
// ---------------------------------------------------------------------------
#define RES    32
#define BATCH  64
#define NPTS   32768               // points per batch
#define GRID   (RES * RES * RES)   // 32768
#define EPS_PD 1e-6f
#define WREG   25.0f

#define TPB        256
#define PPT        8                                  // points per thread
#define NBLOCKS    ((BATCH * NPTS) / (TPB * PPT))     // 1024
#define BLK_PER_B  (NBLOCKS / BATCH)                  // 16
#define PTS_PER_BLK (TPB * PPT)                       // 2048

typedef float v2f __attribute__((ext_vector_type(2)));
typedef float v8f __attribute__((ext_vector_type(8)));

// ---------------------------------------------------------------------------
// ds_swizzle group-of-32 patterns: offset = (and<<10)|(or<<5)|xor, bit15=0
// SWAPXn: and=0x1f, or=0, xor=n
// ---------------------------------------------------------------------------
#define SWZ_SWAPX1 0x7C01
#define SWZ_SWAPX2 0x7C02
#define SWZ_SWAPX4 0x7C04
#define SWZ_SWAPX8 0x7C08

template <int PAT>
__device__ __forceinline__ float swz_add(float v) {
    int o = __builtin_amdgcn_ds_swizzle(__float_as_int(v), PAT);
    return v + __int_as_float(o);
}

// ---------------------------------------------------------------------------
// Wave(32) reduction using V_WMMA_F32_16X16X4_F32.
// A = all-ones (16x4)  =>  D[m][n] = sum_k B[k][n]: every row of D equals the
// column-sum vector of B. Each lane contributes its partial via B-vgpr0 (0 in
// B-vgpr1); summing the 16 columns of a D row gives the 32-lane total,
// independent of the exact K-row->lane mapping. Requires EXEC == all ones.
// ---------------------------------------------------------------------------
__device__ __forceinline__ float wave_reduce_wmma(float v) {
    v2f a;  a[0] = 1.0f; a[1] = 1.0f;     // A: all ones
    v2f bm; bm[0] = v;   bm[1] = 0.0f;    // B: one partial per lane
    v8f c = {};
    c = __builtin_amdgcn_wmma_f32_16x16x4_f32(
            false, a, false, bm, (short)0, c, false, false);
    float s = c[0];                       // lanes 0-15: row0 cols; 16-31: row8 cols (equal)
    // Sum 16 columns within each half-wave (xor bits 0..3) via ds_swizzle.
    s = swz_add<SWZ_SWAPX1>(s);
    s = swz_add<SWZ_SWAPX2>(s);
    s = swz_add<SWZ_SWAPX4>(s);
    s = swz_add<SWZ_SWAPX8>(s);
    return s;                             // full 32-lane total in every lane
}

// ---------------------------------------------------------------------------
// Main kernel: PPT points per thread; 3 plane reflections + grid gathers.
// ---------------------------------------------------------------------------
__global__ void __launch_bounds__(TPB)
sym_loss_kernel(const float* __restrict__ pc,      // (B, N, 3)
                const float* __restrict__ aux,     // (B, G, 3)
                const float* __restrict__ vox,     // (B, G)
                const float* __restrict__ planes,  // (3, B, 4)
                float* __restrict__ partials)      // (NBLOCKS)
{
    const int b    = blockIdx.x / BLK_PER_B;                 // uniform per block
    const int base = b * NPTS + (blockIdx.x % BLK_PER_B) * PTS_PER_BLK;
    const int gbase = b << 15;                               // b * GRID

    // Per-plane constants: uniform per block (scalar loads); the reciprocal is
    // computed once per thread instead of once per point.
    float nx[3], ny[3], nz[3], pd[3], inv[3];
    #pragma unroll
    for (int p = 0; p < 3; ++p) {
        const float* pl = planes + ((size_t)(p * BATCH + b)) * 4;
        nx[p] = pl[0]; ny[p] = pl[1]; nz[p] = pl[2]; pd[p] = pl[3];
        inv[p] = 1.0f / (nx[p] * nx[p] + ny[p] * ny[p] + nz[p] * nz[p]);
    }

    float acc = 0.0f;

    #pragma unroll
    for (int k = 0; k < PPT; ++k) {
        const int gid = base + k * TPB + threadIdx.x;        // coalesced
        const float px = pc[gid * 3 + 0];
        const float py = pc[gid * 3 + 1];
        const float pz = pc[gid * 3 + 2];

        #pragma unroll
        for (int p = 0; p < 3; ++p) {
            // Householder reflection: r = p - 2*t*n,  t = (p.n + d) * inv|n|^2
            const float t  = (px * nx[p] + py * ny[p] + pz * nz[p] + pd[p]) * inv[p];
            const float s2 = 2.0f * t;
            const float rx = px - s2 * nx[p];
            const float ry = py - s2 * ny[p];
            const float rz = pz - s2 * nz[p];

            // Voxel index: truncate-toward-zero (matches astype(int32)), clamp.
            int ix = (int)((rx + 0.5f) * (float)RES);
            int iy = (int)((ry + 0.5f) * (float)RES);
            int iz = (int)((rz + 0.5f) * (float)RES);
            ix = min(max(ix, 0), RES - 1);
            iy = min(max(iy, 0), RES - 1);
            iz = min(max(iz, 0), RES - 1);

            const int g = gbase + (ix << 10) + (iy << 5) + iz;

            // Gathers (batch working set 512 KB; whole aux+vox 33.5 MB -> L2)
            const float v  = vox[g];
            const float* tg = aux + (size_t)g * 3;
            const float dx = rx - tg[0] + EPS_PD;
            const float dy = ry - tg[1] + EPS_PD;
            const float dz = rz - tg[2] + EPS_PD;

            acc += __builtin_amdgcn_sqrtf(dx * dx + dy * dy + dz * dz) * (1.0f - v);
        }
    }

    // ---- block reduction (deterministic) ----
    const float wave_sum = wave_reduce_wmma(acc);

    __shared__ float smem[TPB / 32];
    const int lane = threadIdx.x & 31;
    const int wave = threadIdx.x >> 5;
    if (lane == 0) smem[wave] = wave_sum;
    __syncthreads();

    if (threadIdx.x == 0) {
        float s = 0.0f;
        #pragma unroll
        for (int i = 0; i < TPB / 32; ++i) s += smem[i];
        partials[blockIdx.x] = s;
    }
}

// ---------------------------------------------------------------------------
// Finish kernel: deterministic sum of block partials + plane regularizer.
// result = (sym_total + WREG * reg_total) / B
// ---------------------------------------------------------------------------
__global__ void __launch_bounds__(TPB)
finish_kernel(const float* __restrict__ partials, int nparts,
              const float* __restrict__ planes,   // (3, B, 4)
              float* __restrict__ out)
{
    float v = 0.0f;
    for (int i = threadIdx.x; i < nparts; i += TPB) v += partials[i];

    if (threadIdx.x < BATCH) {
        const int b = threadIdx.x;
        float nv[3][3];
        #pragma unroll
        for (int p = 0; p < 3; ++p) {
            const float* pl = planes + ((size_t)(p * BATCH + b)) * 4;
            const float x = pl[0], y = pl[1], z = pl[2];
            float nrm = sqrtf(x * x + y * y + z * z);
            nrm = fmaxf(nrm, 1e-12f);
            nv[p][0] = x / nrm;
            nv[p][1] = y / nrm;
            nv[p][2] = z / nrm;
        }
        float s = 0.0f;
        #pragma unroll
        for (int i = 0; i < 3; ++i)
            #pragma unroll
            for (int j = 0; j < 3; ++j) {
                const float m = nv[i][j] * nv[j][i] - (i == j ? 1.0f : 0.0f);
                s += m * m;
            }
        v += WREG * s;
    }

    __shared__ float red[TPB];
    red[threadIdx.x] = v;
    __syncthreads();
    #pragma unroll
    for (int off = TPB / 2; off > 0; off >>= 1) {
        if (threadIdx.x < off) red[threadIdx.x] += red[threadIdx.x + off];
        __syncthreads();
    }
    if (threadIdx.x == 0) out[0] = red[0] / (float)BATCH;
}

// ---------------------------------------------------------------------------
// Entry point
// ---------------------------------------------------------------------------
extern "C" void kernel_launch(void* const* d_in, const int* in_sizes, int n_in,
                              void* d_out, int out_size, void* d_ws, size_t ws_size,
                              hipStream_t stream)
{
    (void)in_sizes; (void)n_in; (void)out_size; (void)ws_size;

    const float* pc     = (const float*)d_in[0];   // point_cloud (64,32768,3)
    const float* aux    = (const float*)d_in[1];   // auxiliary_data (64,32768,3)
    const float* vox    = (const float*)d_in[2];   // voxel_data (64,32768,1)
    const float* planes = (const float*)d_in[3];   // planes (3,64,4)
    float*       out    = (float*)d_out;           // scalar fp32
    float*       parts  = (float*)d_ws;            // NBLOCKS floats (4 KB)

    sym_loss_kernel<<<NBLOCKS, TPB, 0, stream>>>(pc, aux, vox, planes, parts);
    finish_kernel<<<1, TPB, 0, stream>>>(parts, NBLOCKS, planes, out);
}